// IcosahedronUnPooling_38654705664296
// MI455X (gfx1250) — compile-verified
//
#include <hip/hip_runtime.h>

typedef float v2f __attribute__((ext_vector_type(2)));
typedef float v8f __attribute__((ext_vector_type(8)));

namespace {
constexpr int kNCoarse = 40962;
constexpr int kMTotal  = 163842;   // 40962 + 122880
constexpr int kCin     = 256;
constexpr int kCout    = 256;
constexpr int BM       = 128;      // output rows per workgroup
constexpr int BN       = 128;      // output cols per workgroup (grid.y = 2)
constexpr int ASTR     = kCin + 4; // padded LDS row stride (16B aligned, bank-spread)
}

// Fused: gather+average unpooling rows on the fly, tiled f32 GEMM via
// V_WMMA_F32_16X16X4_F32, bias add, store. Full fp32 numerics (kernel is
// HBM-write bound; low precision buys nothing here).
__global__ __launch_bounds__(256)
void unpool_linear_wmma_f32(const float* __restrict__ x,
                            const int*   __restrict__ idx,     // [N_NEW][2]
                            const float* __restrict__ Wm,      // [Cout][Cin]
                            const float* __restrict__ bias,    // [Cout]
                            float*       __restrict__ out)     // [M][Cout]
{
  extern __shared__ float lds[];
  float* __restrict__ Alds  = lds;              // [BM][ASTR] row-major
  float* __restrict__ Wtlds = lds + BM * ASTR;  // paired-K: ((k>>1)*BN + n)*2 + (k&1)

  const int tid   = threadIdx.x;
  const int mBase = blockIdx.x * BM;
  const int nBase = blockIdx.y * BN;

  // ---------------- Stage A tile: coarse rows copied, fine rows = avg of 2 parents
  {
    const int rsub = tid >> 6;          // 4 rows per pass
    const int kq   = (tid & 63) << 2;   // float4 column offset
#pragma unroll 4
    for (int pass = 0; pass < BM / 4; ++pass) {
      const int r = pass * 4 + rsub;
      const int m = mBase + r;
      float4 v;
      if (m < kNCoarse) {
        v = *(const float4*)(x + (size_t)m * kCin + kq);
      } else if (m < kMTotal) {
        const int j  = m - kNCoarse;
        const int i0 = idx[2 * j];
        const int i1 = idx[2 * j + 1];
        const float4 a = *(const float4*)(x + (size_t)i0 * kCin + kq);
        const float4 b = *(const float4*)(x + (size_t)i1 * kCin + kq);
        v = make_float4(0.5f * (a.x + b.x), 0.5f * (a.y + b.y),
                        0.5f * (a.z + b.z), 0.5f * (a.w + b.w));
      } else {
        v = make_float4(0.f, 0.f, 0.f, 0.f);   // tail padding rows
      }
      *(float4*)(Alds + r * ASTR + kq) = v;
    }
  }

  // ---------------- Stage W^T tile (transpose W[n][k] -> paired-K LDS layout)
  {
    const int nsub = tid & 63;   // n spreads across lanes -> conflict-free LDS stores
    const int kgrp = tid >> 6;
#pragma unroll 4
    for (int pass = 0; pass < 32; ++pass) {
      const int n  = nsub + ((pass & 1) << 6);           // 0..127
      const int k4 = (kgrp + ((pass >> 1) << 2)) << 2;   // 0..252, step 4 per thread
      const float4 wv = *(const float4*)(Wm + (size_t)(nBase + n) * kCin + k4);
      float* dst = Wtlds + (size_t)(k4 >> 1) * (BN * 2) + n * 2;
      dst[0]          = wv.x;   // k4   (even slot of pair)
      dst[1]          = wv.y;   // k4+1 (odd slot)
      dst[BN * 2 + 0] = wv.z;   // k4+2
      dst[BN * 2 + 1] = wv.w;   // k4+3
    }
  }

  __syncthreads();

  // ---------------- Compute: 8 waves, each owns 32(M) x 64(N) = 2x4 WMMA tiles
  const int lane = tid & 31;
  const int wv   = tid >> 5;
  const int wm   = wv & 3;        // M group of 32 rows
  const int wn   = wv >> 2;       // N group of 64 cols
  const int lr   = lane & 15;
  const int lh   = lane >> 4;     // lane half: selects K pair (A/B) and M half (C)

  v8f acc[2][4] = {};

  const float* Abase0 = Alds + (wm * 32 + lr) * ASTR;        // A frag: lanes=M rows
  const float* Abase1 = Abase0 + 16 * ASTR;
  const float* Bbase  = Wtlds + (wn * 64 + lr) * 2;          // B frag: lanes=N cols

  for (int k0 = 0; k0 < kCin; k0 += 4) {
    const int koff = k0 + (lh << 1);                 // K pair: lanes 0-15 -> k0,k0+1 ; 16-31 -> k0+2,k0+3
    const v2f a0 = *(const v2f*)(Abase0 + koff);
    const v2f a1 = *(const v2f*)(Abase1 + koff);
    const int krow = (koff >> 1) * (BN * 2);
#pragma unroll
    for (int nt = 0; nt < 4; ++nt) {
      const v2f bf = *(const v2f*)(Bbase + krow + nt * 32);  // nt*16 cols * 2 floats
      acc[0][nt] = __builtin_amdgcn_wmma_f32_16x16x4_f32(
          false, a0, false, bf, (short)0, acc[0][nt], false, false);
      acc[1][nt] = __builtin_amdgcn_wmma_f32_16x16x4_f32(
          false, a1, false, bf, (short)0, acc[1][nt], false, false);
    }
  }

  // ---------------- Epilogue: bias add + store (C layout: VGPR e -> M = e + 8*lh, lane -> N)
#pragma unroll
  for (int nt = 0; nt < 4; ++nt) {
    const int col = nBase + wn * 64 + nt * 16 + lr;
    const float bv = bias[col];
#pragma unroll
    for (int mt = 0; mt < 2; ++mt) {
      const int rowBase = mBase + wm * 32 + mt * 16 + (lh << 3);
#pragma unroll
      for (int e = 0; e < 8; ++e) {
        const int row = rowBase + e;
        if (row < kMTotal)
          out[(size_t)row * kCout + col] = acc[mt][nt][e] + bv;
      }
    }
  }
}

extern "C" void kernel_launch(void* const* d_in, const int* in_sizes, int n_in,
                              void* d_out, int out_size, void* d_ws, size_t ws_size,
                              hipStream_t stream) {
  (void)in_sizes; (void)n_in; (void)out_size; (void)d_ws; (void)ws_size;
  const float* x    = (const float*)d_in[0];
  const int*   idx  = (const int*)d_in[1];
  const float* Wm   = (const float*)d_in[2];
  const float* bias = (const float*)d_in[3];
  float* out = (float*)d_out;

  dim3 grid((kMTotal + BM - 1) / BM, kCout / BN, 1);   // 1281 x 2
  const size_t shmem = (size_t)(BM * ASTR + kCin * BN) * sizeof(float); // ~258 KB < 320 KB WGP LDS
  hipLaunchKernelGGL(unpool_linear_wmma_f32, grid, dim3(256, 1, 1), shmem, stream,
                     x, idx, Wm, bias, out);
}